// Fuser_1881195675915
// MI455X (gfx1250) — compile-verified
//
#include <hip/hip_runtime.h>

// CDNA5 / gfx1250, wave32.
// out[b, d] = sum_{t < lengths[b]} knowledge[b, t, d]
// (softmax over a singleton axis == 1.0 -> the projection GEMMs in the
//  reference are algebraically dead; the optimal kernel is a pure HBM-rate
//  masked stream-reduction: ~128-256 MB at 23.3 TB/s ~= 5.5-11 us.
//  The reduction runs on the matrix pipe: D = A x B + C with A = ones/mask.)

typedef __attribute__((ext_vector_type(2))) float v2f;
typedef __attribute__((ext_vector_type(8))) float v8f;

#define NB 16
#define NT 2048
#define KD 2048   // KDIM = 2*HIDDEN

// Each wave owns 64 consecutive d-columns. Per K-step group of 4 rows:
//   load0 = float4 at row (t+half),   cols d0w + 4n .. 4n+3   (b128)
//   load1 = float4 at row (t+2+half), cols d0w + 4n .. 4n+3
// Component j of (load0, load1) is the B operand (4x16) of tile j:
//   VGPR0: lanes0-15 = row K0, lanes16-31 = K1 ; VGPR1 = K2/K3.
// Tile j, N-index n  <->  memory column d0w + 4n + j.
__global__ __launch_bounds__(256) void masked_sum_wmma(
    const float* __restrict__ knowledge,
    const int*   __restrict__ lengths,
    float*       __restrict__ dst,      // per-chunk partials (or out if 1 chunk)
    int tchunk)                          // rows per T-chunk (multiple of 4)
{
    const int b     = blockIdx.x;
    const int d0blk = blockIdx.y * 512;          // 8 waves * 64 cols
    const int chunk = blockIdx.z;
    const int tid   = threadIdx.x;
    const int wave  = tid >> 5;
    const int lane  = tid & 31;
    const int half  = lane >> 4;                 // K-row select within pair
    const int n     = lane & 15;                 // N index inside each tile
    const int d0w   = d0blk + wave * 64;

    const int len    = lengths[b];               // block-uniform
    const int tStart = chunk * tchunk;
    const int tEnd   = tStart + tchunk;
    const int effLen = (len < tEnd) ? len : tEnd;
    const int nRows  = effLen - tStart;          // may be <= 0
    const int tFull  = tStart + ((nRows > 0 ? nRows : 0) & ~3);

    const float* kb = knowledge + (size_t)b * NT * KD;
    const float4* p0 = (const float4*)(kb + (size_t)(tStart + half) * KD + d0w + 4 * n);
    const float4* p1 = (const float4*)(kb + (size_t)(tStart + 2 + half) * KD + d0w + 4 * n);
    const int rowstep4 = KD;                     // 4 rows, in float4 units

    v8f acc0 = {0,0,0,0,0,0,0,0};
    v8f acc1 = {0,0,0,0,0,0,0,0};
    v8f acc2 = {0,0,0,0,0,0,0,0};
    v8f acc3 = {0,0,0,0,0,0,0,0};

    const v2f ones = {1.0f, 1.0f};

    #pragma unroll 2
    for (int t = tStart; t < tFull; t += 4) {
        __builtin_prefetch((const float*)p0 + (size_t)16 * KD, 0, 1);
        __builtin_prefetch((const float*)p1 + (size_t)16 * KD, 0, 1);
        float4 r0 = *p0;                         // rows t+half   (K0/K1)
        float4 r1 = *p1;                         // rows t+2+half (K2/K3)
        p0 += rowstep4; p1 += rowstep4;

        v2f b0 = {r0.x, r1.x}, b1 = {r0.y, r1.y};
        v2f b2 = {r0.z, r1.z}, b3 = {r0.w, r1.w};
        acc0 = __builtin_amdgcn_wmma_f32_16x16x4_f32(false, ones, false, b0, (short)0, acc0, false, false);
        acc1 = __builtin_amdgcn_wmma_f32_16x16x4_f32(false, ones, false, b1, (short)0, acc1, false, false);
        acc2 = __builtin_amdgcn_wmma_f32_16x16x4_f32(false, ones, false, b2, (short)0, acc2, false, false);
        acc3 = __builtin_amdgcn_wmma_f32_16x16x4_f32(false, ones, false, b3, (short)0, acc3, false, false);
    }

    // Masked tail: 1..3 live rows (block-uniform condition; EXEC stays full).
    // Over-read rows stay < NT (chunk starts are 4-aligned); A-weights zero them.
    if (tFull < effLen) {
        v2f a;
        a.x = ((tFull + 2 * half)     < effLen) ? 1.0f : 0.0f;
        a.y = ((tFull + 2 * half + 1) < effLen) ? 1.0f : 0.0f;
        float4 r0 = *p0;
        float4 r1 = *p1;
        v2f b0 = {r0.x, r1.x}, b1 = {r0.y, r1.y};
        v2f b2 = {r0.z, r1.z}, b3 = {r0.w, r1.w};
        acc0 = __builtin_amdgcn_wmma_f32_16x16x4_f32(false, a, false, b0, (short)0, acc0, false, false);
        acc1 = __builtin_amdgcn_wmma_f32_16x16x4_f32(false, a, false, b1, (short)0, acc1, false, false);
        acc2 = __builtin_amdgcn_wmma_f32_16x16x4_f32(false, a, false, b2, (short)0, acc2, false, false);
        acc3 = __builtin_amdgcn_wmma_f32_16x16x4_f32(false, a, false, b3, (short)0, acc3, false, false);
    }

    // All 16 D-rows identical; VGPR0 lanes0-15 = (M=0, N=lane). Column of
    // tile j at N=n is d0w + 4n + j -> one coalesced float4 store per lane.
    if (lane < 16) {
        float4 o = {acc0[0], acc1[0], acc2[0], acc3[0]};
        *(float4*)(dst + (size_t)chunk * NB * KD + (size_t)b * KD + d0w + 4 * lane) = o;
    }
}

// Deterministic cross-chunk reduce: out[i] = sum_c partial[c][i]
__global__ __launch_bounds__(256) void reduce_chunks(
    const float* __restrict__ part, float* __restrict__ out, int nchunk)
{
    int i = (blockIdx.x * 256 + threadIdx.x) * 4;
    if (i < NB * KD) {
        float4 s = *(const float4*)(part + i);
        for (int c = 1; c < nchunk; ++c) {
            float4 p = *(const float4*)(part + (size_t)c * NB * KD + i);
            s.x += p.x; s.y += p.y; s.z += p.z; s.w += p.w;
        }
        *(float4*)(out + i) = s;
    }
}

extern "C" void kernel_launch(void* const* d_in, const int* in_sizes, int n_in,
                              void* d_out, int out_size, void* d_ws, size_t ws_size,
                              hipStream_t stream) {
    // setup_inputs order: knowledge, sentences, lengths, Wk, bk, Ws, bs
    const float* knowledge = (const float*)d_in[0];
    const int*   lengths   = (const int*)  d_in[2];
    float*       out       = (float*)      d_out;
    float*       ws        = (float*)      d_ws;

    // Deterministic T-split: as fine as scratch allows (latency hiding +
    // load balance across ragged lengths). 16 chunks -> 1024 blocks, 8192 waves.
    int tsplit = 1;
    const size_t per = (size_t)NB * KD * sizeof(float);
    if      (ws_size >= 16 * per) tsplit = 16;
    else if (ws_size >=  8 * per) tsplit = 8;
    else if (ws_size >=  4 * per) tsplit = 4;

    if (tsplit > 1) {
        dim3 grid(NB, KD / 512, tsplit);
        masked_sum_wmma<<<grid, 256, 0, stream>>>(knowledge, lengths, ws, NT / tsplit);
        reduce_chunks<<<(NB * KD / 4 + 255) / 256, 256, 0, stream>>>(ws, out, tsplit);
    } else {
        dim3 grid(NB, KD / 512, 1);                   // single-pass fallback
        masked_sum_wmma<<<grid, 256, 0, stream>>>(knowledge, lengths, out, NT);
    }
}